// ComplexScaledDotProductAttention_1365799600287
// MI455X (gfx1250) — compile-verified
//
#include <hip/hip_runtime.h>

typedef __attribute__((ext_vector_type(2))) float v2f;
typedef __attribute__((ext_vector_type(8))) float v8f;

#define L      2048
#define DH     64
#define QB     16            // query rows per workgroup (score strip = QB x L in LDS)
#define STRIDE 2050          // padded LDS row stride in floats (even, avoids bank conflicts)
#define INV_T  0.125f        // 1 / sqrt(d_head = 64)

static_assert(L == 2048 && DH == 64, "decomposition assumes L=2048, d=64");

__device__ __forceinline__ v8f wmma4(v2f a, v2f b, v8f c) {
    return __builtin_amdgcn_wmma_f32_16x16x4_f32(false, a, false, b, (short)0, c,
                                                 false, false);
}

// ---------------------------------------------------------------------------
// Fused complex attention: one workgroup handles 16 query rows of one (b,h).
// Score strip lives entirely in LDS (128 KB < 320 KB/WGP) -> single HBM pass
// over the attention matrix.
// ---------------------------------------------------------------------------
__global__ __launch_bounds__(256) void fused_cattn_kernel(
    const float* __restrict__ qr, const float* __restrict__ qi,
    const float* __restrict__ kr, const float* __restrict__ ki,
    const float* __restrict__ vr, const float* __restrict__ vi,
    float* __restrict__ out, float* __restrict__ attn)
{
    extern __shared__ float lds[];              // [QB][STRIDE] strip, then red[256], invsum[QB]
    float* red    = lds + QB * STRIDE;
    float* invsum = red + 256;

    const int lane    = threadIdx.x & 31;
    const int wave    = threadIdx.x >> 5;
    const int qblk    = blockIdx.x & 127;       // L/QB = 128 query blocks per bh
    const int bh      = blockIdx.x >> 7;
    const int qBase   = qblk * QB;

    const int ln      = lane & 15;              // A-row / B-col within a 16x16 tile
    const int koff    = (lane >> 4) << 1;       // K sub-offset: 0 or 2
    const int rowHalf = (lane >> 4) << 3;       // C/D row half: 0 or 8

    // ---- Phase 1: E[q,k] = exp(|complex QK|/T) -> LDS strip --------------
    {
        const float* Qr = qr + ((size_t)bh * L + qBase + ln) * DH;
        const float* Qi = qi + ((size_t)bh * L + qBase + ln) * DH;

        // Preload all Q fragments once; reused across the wave's 16 k-tiles.
        v2f aqr[16], aqi[16];
#pragma unroll
        for (int s = 0; s < 16; ++s) {
            aqr[s] = *(const v2f*)(Qr + s * 4 + koff);
            aqi[s] = *(const v2f*)(Qi + s * 4 + koff);
        }

#pragma unroll 1
        for (int j = 0; j < 16; ++j) {
            const int kt = wave * 16 + j;       // this wave's k-tile
            const float* Kr = kr + ((size_t)bh * L + kt * 16 + ln) * DH;
            const float* Ki = ki + ((size_t)bh * L + kt * 16 + ln) * DH;

            v8f p1 = {};   // qr.kr + qi.ki  -> s_real
            v8f p3 = {};   // qi.kr
            v8f p4 = {};   // qr.ki          -> s_imag = p3 - p4
#pragma unroll
            for (int s = 0; s < 16; ++s) {
                v2f br = *(const v2f*)(Kr + s * 4 + koff);
                v2f bi = *(const v2f*)(Ki + s * 4 + koff);
                p1 = wmma4(aqr[s], br, p1);
                p1 = wmma4(aqi[s], bi, p1);
                p3 = wmma4(aqi[s], br, p3);
                p4 = wmma4(aqr[s], bi, p4);
            }
            const int col = kt * 16 + ln;
#pragma unroll
            for (int v = 0; v < 8; ++v) {
                float sr = p1[v];
                float si = p3[v] - p4[v];
                lds[(rowHalf + v) * STRIDE + col] =
                    __expf(sqrtf(sr * sr + si * si) * INV_T);
            }
        }
    }
    __syncthreads();

    // ---- Phase 2: deterministic row sums -> invsum[QB] --------------------
    {
        const int t   = threadIdx.x;
        const int r   = t & 15;
        const int seg = t >> 4;                 // 16 segments of 128 columns
        float s = 0.f;
        const float* p = lds + r * STRIDE + seg * 128;
#pragma unroll 8
        for (int c2 = 0; c2 < 128; ++c2) s += p[c2];
        red[t] = s;
        __syncthreads();
        if (t < 16) {
            float tot = 0.f;
#pragma unroll
            for (int sg = 0; sg < 16; ++sg) tot += red[sg * 16 + t];
            invsum[t] = 1.0f / tot;
        }
        __syncthreads();
    }

    // ---- Phase 3: stream normalized attn to HBM (write-once, NT) ----------
    {
        const int t = threadIdx.x;
        float* arow = attn + ((size_t)bh * L + qBase) * L;
#pragma unroll 4
        for (int i = 0; i < 128; ++i) {         // QB*L / 256 elements per thread
            const int row = i >> 3;             // 8 iterations span one row
            const int col = ((i & 7) << 8) + t; // coalesced across the block
            float v = lds[row * STRIDE + col] * invsum[row];
            __builtin_nontemporal_store(v, arow + (size_t)row * L + col);
        }
    }

    // ---- Phase 4: PV from LDS strip; wave = (d-tile, real/imag) ------------
    {
        const int dt = wave >> 1;               // 4 d-tiles of 16
        const int c  = wave & 1;                // 0 = real, 1 = imag
        const float* V = (c ? vi : vr) + (size_t)bh * L * DH + dt * 16 + ln;
        const float inv = invsum[ln];           // this lane's A-row scale
        const float* aRow = lds + (size_t)ln * STRIDE;

        v8f acc = {};
#pragma unroll 4
        for (int k0 = 0; k0 < L; k0 += 4) {
            v2f a = *(const v2f*)(aRow + k0 + koff);   // ds_load_b64, padded stride
            a.x *= inv;
            a.y *= inv;
            v2f b;
            b.x = V[(size_t)(k0 + koff) * DH];
            b.y = V[(size_t)(k0 + koff + 1) * DH];
            acc = wmma4(a, b, acc);
        }

        float* o = out + (((size_t)bh * L + qBase) * DH + dt * 16 + ln) * 2 + c;
#pragma unroll
        for (int v = 0; v < 8; ++v)
            o[(size_t)(rowHalf + v) * DH * 2] = acc[v];
    }
}

// ---------------------------------------------------------------------------
extern "C" void kernel_launch(void* const* d_in, const int* in_sizes, int n_in,
                              void* d_out, int out_size, void* d_ws, size_t ws_size,
                              hipStream_t stream) {
    const float* qr = (const float*)d_in[0];
    const float* qi = (const float*)d_in[1];
    const float* kr = (const float*)d_in[2];
    const float* ki = (const float*)d_in[3];
    const float* vr = (const float*)d_in[4];
    const float* vi = (const float*)d_in[5];
    float* out = (float*)d_out;

    const int BH = in_sizes[0] / (L * DH);               // 4*16 = 64
    float* attn = out + (size_t)BH * L * DH * 2;         // attn follows output

    const int blocks = BH * (L / QB);                    // 64 * 128 = 8192
    const size_t shmem = (size_t)(QB * STRIDE + 256 + QB) * sizeof(float); // ~129.2 KB

    fused_cattn_kernel<<<blocks, 256, shmem, stream>>>(qr, qi, kr, ki, vr, vi,
                                                       out, attn);
}